// GCN_12893491823230
// MI455X (gfx1250) — compile-verified
//
#include <hip/hip_runtime.h>
#include <hip/hip_bf16.h>
#include <math.h>

// ---------------------------------------------------------------------------
// 2-layer GCN for MI455X (gfx1250, wave32).
//   layer: out = D^-1/2 (A+I) D^-1/2 (x @ W) + b ; sigmoid
// GEMM uses full-precision fp32 WMMA (v_wmma_f32_16x16x4_f32) since the
// reference is fp32. Edge aggregation = CH/4-lanes-per-edge float4 gather (L2-resident
// xw table, 51MB << 192MB L2) + global_atomic_add_f32 scatter.
// ---------------------------------------------------------------------------

typedef __attribute__((ext_vector_type(2))) float v2f;
typedef __attribute__((ext_vector_type(8))) float v8f;

#define GCN_IN_CH 128

// ---- degree / norm ---------------------------------------------------------
__global__ void k_deg_init(float* __restrict__ deg, int n) {
    int i = blockIdx.x * blockDim.x + threadIdx.x;
    if (i < n) deg[i] = 1.0f;                      // self-loop contributes 1
}

__global__ void k_deg_accum(const int* __restrict__ dst, float* __restrict__ deg, int n_edges) {
    int e = blockIdx.x * blockDim.x + threadIdx.x;
    if (e < n_edges) atomicAdd(&deg[dst[e]], 1.0f);
}

__global__ void k_deg_to_dinv(float* __restrict__ deg, int n) {
    int i = blockIdx.x * blockDim.x + threadIdx.x;
    if (i < n) deg[i] = rsqrtf(deg[i]);            // deg >= 1 always
}

// ---- fp32 WMMA GEMM:  Y[n x NOUT] = X[n x 128] @ W[128 x NOUT] -------------
// One wave per 16-row tile; the A fragment is reused across all N-tiles.
// A layout (16x4 f32): lane L holds row M=L&15, K = {0,1}+2*(L>=16).
// B layout (4x16 f32): VGPR v holds K = v + 2*(L>=16), N = L&15.
// C/D layout: VGPR v holds row M = v + 8*(L>=16), N = L&15.
template <int NOUT>
__global__ __launch_bounds__(256) void k_gemm_wmma(const float* __restrict__ X,
                                                   const float* __restrict__ W,
                                                   float* __restrict__ Y, int ntiles) {
    constexpr int NT = NOUT / 16;
    const int wave = threadIdx.x >> 5;
    const int lane = threadIdx.x & 31;
    const int tile = blockIdx.x * 8 + wave;
    if (tile >= ntiles) return;                    // wave-uniform exit

    const int nlo = lane & 15;
    const int hi  = lane >> 4;                     // 0 or 1 -> K offset 0 or 2
    const float* xrow = X + (size_t)(tile * 16 + nlo) * GCN_IN_CH + 2 * hi;

    v8f acc[NT] = {};
#pragma unroll 4
    for (int kk = 0; kk < GCN_IN_CH / 4; ++kk) {
        const v2f a = *(const v2f*)(xrow + 4 * kk);            // contiguous K pair
        const float* wp = W + (size_t)(4 * kk + 2 * hi) * NOUT + nlo;
#pragma unroll
        for (int nt = 0; nt < NT; ++nt) {
            v2f b;
            b.x = wp[nt * 16];                                  // K row k
            b.y = wp[nt * 16 + NOUT];                           // K row k+1
            acc[nt] = __builtin_amdgcn_wmma_f32_16x16x4_f32(
                false, a, false, b, (short)0, acc[nt], false, false);
        }
    }

    const int row0 = tile * 16 + 8 * hi;
#pragma unroll
    for (int nt = 0; nt < NT; ++nt) {
#pragma unroll
        for (int v = 0; v < 8; ++v)
            Y[(size_t)(row0 + v) * NOUT + nt * 16 + nlo] = acc[nt][v];
    }
}

// scalar tail for rows beyond the last full 16-row tile (none for N=100000)
template <int NOUT>
__global__ void k_gemm_tail(const float* __restrict__ X, const float* __restrict__ W,
                            float* __restrict__ Y, int row0, int n) {
    int idx = blockIdx.x * blockDim.x + threadIdx.x;
    int row = row0 + idx / NOUT;
    int col = idx % NOUT;
    if (row >= n) return;
    float s = 0.f;
    for (int k = 0; k < GCN_IN_CH; ++k)
        s += X[(size_t)row * GCN_IN_CH + k] * W[(size_t)k * NOUT + col];
    Y[(size_t)row * NOUT + col] = s;
}

// ---- misc elementwise ------------------------------------------------------
__global__ void k_zero(float* __restrict__ p, size_t total) {
    size_t i = blockIdx.x * (size_t)blockDim.x + threadIdx.x;
    if (i < total) p[i] = 0.0f;
}

template <int CH>
__global__ void k_bias_sigmoid(float* __restrict__ y, const float* __restrict__ b, size_t total) {
    size_t i = blockIdx.x * (size_t)blockDim.x + threadIdx.x;
    if (i >= total) return;
    float v = y[i] + b[i % CH];
    y[i] = 1.0f / (1.0f + expf(-v));
}

// ---- edge aggregation: out[dst] += xw[src] * dinv[src]*dinv[dst] -----------
// CH/4 lanes cooperate on one edge; float4 gather (L2-resident), 4 f32 atomics.
// Edges e in [0, E) are real; e in [E, E+N) are self-loops (src=dst=e-E).
template <int CH>
__global__ __launch_bounds__(256) void k_scatter(const float* __restrict__ xw,
                                                 const int* __restrict__ src,
                                                 const int* __restrict__ dst,
                                                 const float* __restrict__ dinv,
                                                 float* __restrict__ out,
                                                 int n_edges, int n_nodes) {
    constexpr int LPE = CH / 4;                    // lanes per edge
    long long t = blockIdx.x * (long long)blockDim.x + threadIdx.x;
    int e = (int)(t / LPE);
    int c = (int)(t % LPE) * 4;
    if (e >= n_edges + n_nodes) return;
    int s, d;
    if (e < n_edges) { s = src[e]; d = dst[e]; }
    else             { s = d = e - n_edges; }
    const float norm = dinv[s] * dinv[d];
    const float4 v = *(const float4*)&xw[(size_t)s * CH + c];
    float* o = &out[(size_t)d * CH + c];
    atomicAdd(o + 0, v.x * norm);
    atomicAdd(o + 1, v.y * norm);
    atomicAdd(o + 2, v.z * norm);
    atomicAdd(o + 3, v.w * norm);
}

// ---------------------------------------------------------------------------
static inline unsigned blks(long long total, int bs) { return (unsigned)((total + bs - 1) / bs); }

extern "C" void kernel_launch(void* const* d_in, const int* in_sizes, int n_in,
                              void* d_out, int out_size, void* d_ws, size_t ws_size,
                              hipStream_t stream) {
    const float* x  = (const float*)d_in[0];
    const int*   ei = (const int*)d_in[1];
    const float* W1 = (const float*)d_in[2];
    const float* b1 = (const float*)d_in[3];
    const float* W2 = (const float*)d_in[4];
    const float* b2 = (const float*)d_in[5];

    const int n = in_sizes[0] / GCN_IN_CH;         // 100000
    const int E = in_sizes[1] / 2;                 // 3200000
    const int* src = ei;                           // edge_index[0]
    const int* dst = ei + E;                       // edge_index[1]

    // workspace: [dinv n][bufXW n*128][bufAgg n*128]   (~103 MB)
    char* ws = (char*)d_ws;
    float* dinv = (float*)ws;
    size_t off = (((size_t)n * 4) + 255) & ~(size_t)255;
    float* bufXW  = (float*)(ws + off);
    float* bufAgg = (float*)(ws + off + (size_t)n * GCN_IN_CH * 4);
    float* out = (float*)d_out;

    const int ntiles = n / 16;
    const int rem = n - ntiles * 16;

    // --- normalization coefficients ---
    k_deg_init<<<blks(n, 256), 256, 0, stream>>>(dinv, n);
    k_deg_accum<<<blks(E, 256), 256, 0, stream>>>(dst, dinv, E);
    k_deg_to_dinv<<<blks(n, 256), 256, 0, stream>>>(dinv, n);

    // --- layer 1: xw1 = x @ W1 (fp32 WMMA), aggregate, sigmoid(+b1) ---
    k_gemm_wmma<128><<<blks(ntiles, 8), 256, 0, stream>>>(x, W1, bufXW, ntiles);
    if (rem > 0)
        k_gemm_tail<128><<<blks((long long)rem * 128, 256), 256, 0, stream>>>(x, W1, bufXW, ntiles * 16, n);

    size_t tot1 = (size_t)n * 128;
    k_zero<<<blks((long long)tot1, 256), 256, 0, stream>>>(bufAgg, tot1);
    long long th1 = (long long)(E + n) * (128 / 4);
    k_scatter<128><<<blks(th1, 256), 256, 0, stream>>>(bufXW, src, dst, dinv, bufAgg, E, n);
    k_bias_sigmoid<128><<<blks((long long)tot1, 256), 256, 0, stream>>>(bufAgg, b1, tot1);
    // bufAgg now holds h (layer-1 activations)

    // --- layer 2: xw2 = h @ W2 (fp32 WMMA), aggregate into d_out, sigmoid(+b2) ---
    k_gemm_wmma<64><<<blks(ntiles, 8), 256, 0, stream>>>(bufAgg, W2, bufXW, ntiles);
    if (rem > 0)
        k_gemm_tail<64><<<blks((long long)rem * 64, 256), 256, 0, stream>>>(bufAgg, W2, bufXW, ntiles * 16, n);

    size_t tot2 = (size_t)n * 64;
    k_zero<<<blks((long long)tot2, 256), 256, 0, stream>>>(out, tot2);
    long long th2 = (long long)(E + n) * (64 / 4);
    k_scatter<64><<<blks(th2, 256), 256, 0, stream>>>(bufXW, src, dst, dinv, out, E, n);
    k_bias_sigmoid<64><<<blks((long long)tot2, 256), 256, 0, stream>>>(out, b2, tot2);
}